// OutputPPBlock_30356828848536
// MI455X (gfx1250) — compile-verified
//
#include <hip/hip_runtime.h>

// MI455X / gfx1250, wave32. fp32 pipeline using V_WMMA_F32_16X16X4_F32.
//
// Phase 1 (memory-bound, ~865 MB @ 23.3 TB/s -> ~37 us floor): fused edge
//   gate + scatter-add via global_atomic_add_f32 into an L2-resident [N,128]
//   accumulator (25.6 MB << 192 MB L2).
// Phase 2 (compute, ~23.6 GFLOP fp32): fully fused node MLP on WMMA.
//   Block = 512 thr = 16 waves, 64-row tile; activations live in LDS;
//   weights staged in double-buffered K=16 slices; 4 C-tiles per wave
//   (32 acc VGPRs -> no spills); global prefetch issued before the WMMA
//   burst so s_wait_loadcnt hides under 16 WMMAs.

typedef float v2f __attribute__((ext_vector_type(2)));
typedef float v8f __attribute__((ext_vector_type(8)));

static constexpr int Hdim = 128;   // edge/hidden width
static constexpr int Ddim = 256;   // MLP width
static constexpr int MT   = 64;    // node rows per block
static constexpr int AS   = 260;   // A-tile LDS stride (260%64==4 -> conflict-free)
static constexpr int BS   = 20;    // B-slice LDS stride (20m mod 64 distinct, 16B aligned)
static constexpr int KS   = 16;    // K depth per staged slice
static constexpr int BBUF = Ddim * BS;  // floats per B buffer (5120)

// ---------------------------------------------------------------- zero
__global__ __launch_bounds__(256) void zero_kernel(float4* __restrict__ p, long long n4) {
  long long i = (long long)blockIdx.x * blockDim.x + threadIdx.x;
  if (i < n4) p[i] = make_float4(0.f, 0.f, 0.f, 0.f);
}

// ---------------------------------------------------------------- edges
// One wave (32 lanes) per edge; lane handles 4 channels (float4 of x).
__global__ __launch_bounds__(256) void edge_scatter_kernel(
    const float* __restrict__ x, const float* __restrict__ rbf,
    const int* __restrict__ idx, const float* __restrict__ W_rbf,
    float* __restrict__ hnode, int E) {
  __shared__ float Wl[Hdim * 6];   // 3 KB
  const int tid = threadIdx.x;
  for (int k = tid; k < Hdim * 6; k += blockDim.x) Wl[k] = W_rbf[k];
  __syncthreads();

  const int lane = tid & 31;
  const int e    = blockIdx.x * 8 + (tid >> 5);
  if (e >= E) return;

  float rv = 0.f;
  if (lane < 6) rv = rbf[(size_t)e * 6 + lane];
  const float r0 = __shfl(rv, 0), r1 = __shfl(rv, 1), r2 = __shfl(rv, 2);
  const float r3 = __shfl(rv, 3), r4 = __shfl(rv, 4), r5 = __shfl(rv, 5);

  const float4 xv = *(const float4*)(x + (size_t)e * Hdim + lane * 4);
  const float xs[4] = {xv.x, xv.y, xv.z, xv.w};
  float* dst = hnode + (size_t)idx[e] * Hdim + lane * 4;

#pragma unroll
  for (int q = 0; q < 4; ++q) {
    const float* wr = Wl + (lane * 4 + q) * 6;
    const float g = r0*wr[0] + r1*wr[1] + r2*wr[2] + r3*wr[3] + r4*wr[4] + r5*wr[5];
    atomicAdd(dst + q, g * xs[q]);      // global_atomic_add_f32, L2-resident target
  }
}

// ---------------------------------------------------------------- node MLP
// 16 waves: rt = w&3 (16-row tile), cg = w>>2 (64-col group = 4 C-tiles).
__device__ __forceinline__ void block_gemm(const float* __restrict__ Wg, int Kd,
                                           const float* __restrict__ A,
                                           float* __restrict__ Bs,
                                           v8f acc[4], int tid, int lane,
                                           int rt, int cg) {
  const int hi = lane >> 4;
  const int m  = lane & 15;
#pragma unroll
  for (int t = 0; t < 4; ++t) acc[t] = v8f{0.f,0.f,0.f,0.f,0.f,0.f,0.f,0.f};

  // Thread's two staging chunks: f and f+512; j = row of W, kk = k offset.
  const int f0 = tid,        j0 = f0 >> 2, kk0 = (f0 & 3) * 4;
  const int f1 = tid + 512,  j1 = f1 >> 2, kk1 = (f1 & 3) * 4;

  const int nstages = Kd >> 4;   // KS = 16
  // Preload stage 0 into buffer 0 (coalesced b128 loads).
  *(float4*)(Bs + j0 * BS + kk0) = *(const float4*)(Wg + j0 * Kd + kk0);
  *(float4*)(Bs + j1 * BS + kk1) = *(const float4*)(Wg + j1 * Kd + kk1);
  __syncthreads();

  for (int st = 0; st < nstages; ++st) {
    float4 pre0, pre1;
    const bool more = (st + 1) < nstages;
    if (more) {  // issue global loads first; wait hides under 16 WMMAs below
      pre0 = *(const float4*)(Wg + j0 * Kd + (st + 1) * KS + kk0);
      pre1 = *(const float4*)(Wg + j1 * Kd + (st + 1) * KS + kk1);
    }

    const float* buf = Bs + (st & 1) * BBUF;
    const float* arow = A + (rt * 16 + m) * AS + st * KS + hi * 2;
#pragma unroll
    for (int ks = 0; ks < 4; ++ks) {
      const v2f a = *(const v2f*)(arow + ks * 4);
#pragma unroll
      for (int ct = 0; ct < 4; ++ct) {
        const int col = cg * 64 + ct * 16 + m;
        const v2f b = *(const v2f*)(buf + col * BS + ks * 4 + hi * 2);
        acc[ct] = __builtin_amdgcn_wmma_f32_16x16x4_f32(
            false, a, false, b, (short)0, acc[ct], false, false);
      }
    }

    if (more) {
      float* nbuf = Bs + ((st + 1) & 1) * BBUF;
      *(float4*)(nbuf + j0 * BS + kk0) = pre0;
      *(float4*)(nbuf + j1 * BS + kk1) = pre1;
    }
    __syncthreads();
  }
}

template <bool SILU>
__device__ __forceinline__ void store_act(const v8f acc[4],
                                          const float* __restrict__ bias,
                                          float* __restrict__ A, int lane,
                                          int rt, int cg) {
  const int hi = lane >> 4;
  const int m  = lane & 15;
#pragma unroll
  for (int ct = 0; ct < 4; ++ct) {
    const int col = cg * 64 + ct * 16 + m;
    const float bv = bias[col];
#pragma unroll
    for (int v = 0; v < 8; ++v) {
      const int row = rt * 16 + hi * 8 + v;   // C layout: VGPR v -> rows v / v+8
      float val = acc[ct][v] + bv;
      if (SILU) val = val / (1.f + __expf(-val));
      A[row * AS + col] = val;
    }
  }
}

__global__ __launch_bounds__(512, 1) void node_mlp_kernel(
    const float* __restrict__ hnode, const float* __restrict__ W_up,
    const float* __restrict__ b_up, const float* __restrict__ Ws,
    const float* __restrict__ bs, const float* __restrict__ W_out,
    float* __restrict__ out, int N) {
  extern __shared__ float smem[];
  float* A  = smem;             // MT x AS activations
  float* Bs = smem + MT * AS;   // 2 x (256 x BS) weight slices / scratch

  const int tid  = threadIdx.x;
  const int lane = tid & 31;
  const int w    = tid >> 5;
  const int rt   = w & 3;
  const int cg   = w >> 2;
  const int base = blockIdx.x * MT;

  // Stage hnode tile [MT x 128] -> LDS (coalesced float4), zero-pad tail rows.
  for (int f = tid; f < MT * (Hdim / 4); f += blockDim.x) {
    const int r  = f >> 5;        // Hdim/4 == 32
    const int c4 = f & 31;
    float4 v = make_float4(0.f, 0.f, 0.f, 0.f);
    if (base + r < N) v = *(const float4*)(hnode + (size_t)(base + r) * Hdim + c4 * 4);
    *(float4*)(A + r * AS + c4 * 4) = v;
  }
  __syncthreads();

  v8f acc[4];

  // Up-projection: [MT,128] @ W_up.T + b_up (no activation)
  block_gemm(W_up, Hdim, A, Bs, acc, tid, lane, rt, cg);
  store_act<false>(acc, b_up, A, lane, rt, cg);
  __syncthreads();

  // 3 x SiLU MLP layers: [MT,256] @ Ws[l].T + bs[l]
  for (int l = 0; l < 3; ++l) {
    block_gemm(Ws + (size_t)l * Ddim * Ddim, Ddim, A, Bs, acc, tid, lane, rt, cg);
    store_act<true>(acc, bs + l * Ddim, A, lane, rt, cg);
    __syncthreads();
  }

  // Final projection to O=1: out[n] = dot(A[n,:], W_out); 8 partials per row.
  const int r    = tid & 63;
  const int part = tid >> 6;          // 0..7
  float p = 0.f;
  for (int j = part * 32; j < part * 32 + 32; ++j)
    p += A[r * AS + j] * W_out[j];
  Bs[part * 64 + r] = p;
  __syncthreads();
  if (tid < 64) {
    float s = 0.f;
#pragma unroll
    for (int q = 0; q < 8; ++q) s += Bs[q * 64 + tid];
    if (base + tid < N) out[base + tid] = s;
  }
}

// ---------------------------------------------------------------- launch
extern "C" void kernel_launch(void* const* d_in, const int* in_sizes, int n_in,
                              void* d_out, int out_size, void* d_ws, size_t ws_size,
                              hipStream_t stream) {
  const float* x     = (const float*)d_in[0];
  const float* rbf   = (const float*)d_in[1];
  const int*   idx   = (const int*)d_in[2];
  const float* W_rbf = (const float*)d_in[3];
  const float* W_up  = (const float*)d_in[4];
  const float* b_up  = (const float*)d_in[5];
  const float* Ws    = (const float*)d_in[6];
  const float* bs    = (const float*)d_in[7];
  const float* W_out = (const float*)d_in[8];

  const int E = in_sizes[0] / Hdim;   // 1.6M
  const int N = out_size;             // O == 1 -> out_size == N
  float* hnode = (float*)d_ws;        // [N,128] accumulator (25.6 MB, L2-resident)

  const long long n4 = (long long)N * Hdim / 4;
  zero_kernel<<<(int)((n4 + 255) / 256), 256, 0, stream>>>((float4*)hnode, n4);

  edge_scatter_kernel<<<(E + 7) / 8, 256, 0, stream>>>(x, rbf, idx, W_rbf, hnode, E);

  const int nblocks = (N + MT - 1) / MT;
  const size_t smem = (size_t)(MT * AS + 2 * BBUF) * sizeof(float);  // ~106.6 KB
  node_mlp_kernel<<<nblocks, 512, smem, stream>>>(hnode, W_up, b_up, Ws, bs, W_out,
                                                  (float*)d_out, N);
}